// TorchPDPostprocess_19997367730696
// MI455X (gfx1250) — compile-verified
//
#include <hip/hip_runtime.h>
#include <hip/hip_bf16.h>
#include <math.h>
#include <stdint.h>

// ---------------------------------------------------------------------------
// PD postprocess for MI455X (gfx1250, wave32).
//   prep:   sigmoid scores + dets = y/192 + anchors@P via V_WMMA_F32_16X16X4_F32
//   sort:   one-workgroup bitonic sort (8192) by score desc, LDS resident
//   mask:   wave32 bitmask NMS (32-col blocks == 1 bit/lane), 256x256 grid
//   scan:   single-WAVE sequential suppression; mask rows prefetched with
//           GLOBAL_LOAD_ASYNC_TO_LDS_B128 (ASYNCcnt) into a double buffer so
//           the 8192 serial steps never stall on raw global-load latency.
//   gather: top-100 rows -> [s, cx, cy, w, kp0x, kp0y, kp2x, kp2y]
// Everything is L2-resident (~9.3 MB ws); compute/latency-shaped, not HBM.
// ---------------------------------------------------------------------------

typedef __attribute__((ext_vector_type(2))) float v2f;
typedef __attribute__((ext_vector_type(8))) float v8f;

#define NDET   8192
#define NCOLB  256          // 8192 / 32 column blocks
#define TOPK   100
#define INV_LEN (1.0f / 192.0f)
#define IOU_THR 0.3f

// Constant P matrix column patterns for cols 0..15 (row0 / row1 of P):
//   row0 has 1 at cols {0,4,6,8,10,12,14} -> 0x5551
//   row1 has 1 at cols {1,5,7,9,11,13,15} -> 0xAAA2
// Cols 16,17 (handled scalar): +ax, +ay.

__global__ void pd_prep_kernel(const float* __restrict__ x,
                               const float* __restrict__ y,
                               const float* __restrict__ anchors,
                               float* __restrict__ dets,
                               float* __restrict__ boxes,
                               float* __restrict__ scores) {
  const int lane = threadIdx.x;        // 0..31, one wave per block
  const int R    = blockIdx.x * 16;    // 16 detection rows per wave

  // A matrix (16x4 f32): lanes 0-15 hold K=0 (vgpr0) / K=1 (vgpr1);
  // lanes 16-31 hold K=2/K=3 which we zero-pad (K is really 2).
  v2f a;
  if (lane < 16) {
    a[0] = anchors[(R + lane) * 2 + 0];
    a[1] = anchors[(R + lane) * 2 + 1];
  } else {
    a[0] = 0.0f; a[1] = 0.0f;
  }

  // B matrix (4x16 f32): row K in matching half/vgpr slot; rows 2,3 are zero
  // (their contribution is killed by the zero A pad regardless).
  const int col = lane & 15;
  v2f b;
  b[0] = (lane < 16) ? (float)((0x5551u >> col) & 1u) : 0.0f;  // K=0 row
  b[1] = (lane < 16) ? (float)((0xAAA2u >> col) & 1u) : 0.0f;  // K=1 row

  // C/D (16x16 f32, 8 VGPRs): lane half selects M base (0 or 8), N = lane&15.
  const int n  = lane & 15;
  const int mb = (lane >> 4) * 8;
  v8f c;
#pragma unroll
  for (int r = 0; r < 8; ++r)
    c[r] = y[(R + mb + r) * 18 + n] * INV_LEN;

  // dets[:, 0:16] = y/192 + anchors @ P   (exact f32, matches reference)
  v8f d = __builtin_amdgcn_wmma_f32_16x16x4_f32(
      /*neg_a=*/false, a, /*neg_b=*/false, b,
      /*c_mod=*/(short)0, c, /*reuse_a=*/false, /*reuse_b=*/false);

#pragma unroll
  for (int r = 0; r < 8; ++r)
    dets[(R + mb + r) * 18 + n] = d[r];

  // Per-row scalar tail: cols 16,17, boxes, sigmoid score.
  if (lane < 16) {
    const int m = R + lane;
    const float ax = anchors[2 * m], ay = anchors[2 * m + 1];
    dets[m * 18 + 16] = y[m * 18 + 16] * INV_LEN + ax;
    dets[m * 18 + 17] = y[m * 18 + 17] * INV_LEN + ay;

    const float cx = y[m * 18 + 0] * INV_LEN + ax;
    const float cy = y[m * 18 + 1] * INV_LEN + ay;
    const float wh = y[m * 18 + 2] * INV_LEN * 0.5f;
    const float hh = y[m * 18 + 3] * INV_LEN * 0.5f;
    boxes[m * 4 + 0] = cx - wh;
    boxes[m * 4 + 1] = cy - hh;
    boxes[m * 4 + 2] = cx + wh;
    boxes[m * 4 + 3] = cy + hh;

    scores[m] = 1.0f / (1.0f + expf(-x[m]));
  }
}

// One workgroup, 1024 threads, 64KB dynamic LDS: bitonic sort (score desc).
__global__ void pd_sort_kernel(const float* __restrict__ scores,
                               const float* __restrict__ boxes,
                               int* __restrict__ sidx,
                               float* __restrict__ sboxes) {
  extern __shared__ char smem[];
  float* ss = (float*)smem;              // 8192 floats
  int*   si = (int*)(ss + NDET);         // 8192 ints
  const int t = threadIdx.x;             // 0..1023

  for (int i = t; i < NDET; i += 1024) { ss[i] = scores[i]; si[i] = i; }
  __syncthreads();

  for (int k = 2; k <= NDET; k <<= 1) {
    for (int j = k >> 1; j > 0; j >>= 1) {
      for (int i = t; i < NDET; i += 1024) {
        const int ixj = i ^ j;
        if (ixj > i) {
          const bool desc = ((i & k) == 0);     // descending overall
          const float av = ss[i], bv = ss[ixj];
          const bool sw = desc ? (av < bv) : (av > bv);
          if (sw) {
            ss[i] = bv; ss[ixj] = av;
            const int tmp = si[i]; si[i] = si[ixj]; si[ixj] = tmp;
          }
        }
      }
      __syncthreads();
    }
  }

  for (int i = t; i < NDET; i += 1024) {
    const int oi = si[i];
    sidx[i] = oi;
    sboxes[i * 4 + 0] = boxes[oi * 4 + 0];
    sboxes[i * 4 + 1] = boxes[oi * 4 + 1];
    sboxes[i * 4 + 2] = boxes[oi * 4 + 2];
    sboxes[i * 4 + 3] = boxes[oi * 4 + 3];
  }
}

// Wave32 bitmask NMS: block = 1 wave; bit jj of mask[row][cb] set iff
// IoU(row, cb*32+jj) > thr and (cb*32+jj) > row  (sorted-domain indices).
__global__ void pd_nms_mask_kernel(const float* __restrict__ sboxes,
                                   unsigned* __restrict__ mask) {
  const int cb = blockIdx.x, rb = blockIdx.y, t = threadIdx.x;
  const int row = rb * 32 + t;
  if (cb < rb) { mask[row * NCOLB + cb] = 0u; return; }  // deterministic zero

  __shared__ float bx1[32], by1[32], bx2[32], by2[32], bar[32];
  const int j = cb * 32 + t;
  const float jx1 = sboxes[j * 4 + 0], jy1 = sboxes[j * 4 + 1];
  const float jx2 = sboxes[j * 4 + 2], jy2 = sboxes[j * 4 + 3];
  bx1[t] = jx1; by1[t] = jy1; bx2[t] = jx2; by2[t] = jy2;
  bar[t] = (jx2 - jx1) * (jy2 - jy1);
  __syncthreads();

  const float rx1 = sboxes[row * 4 + 0], ry1 = sboxes[row * 4 + 1];
  const float rx2 = sboxes[row * 4 + 2], ry2 = sboxes[row * 4 + 3];
  const float rar = (rx2 - rx1) * (ry2 - ry1);

  unsigned m = 0u;
#pragma unroll
  for (int jj = 0; jj < 32; ++jj) {
    const int jg = cb * 32 + jj;
    if (jg <= row) continue;
    const float iw = fmaxf(fminf(rx2, bx2[jj]) - fmaxf(rx1, bx1[jj]), 0.0f);
    const float ih = fmaxf(fminf(ry2, by2[jj]) - fmaxf(ry1, by1[jj]), 0.0f);
    const float inter = iw * ih;
    const float iou = inter / (rar + bar[jj] - inter);
    if (iou > IOU_THR) m |= (1u << jj);
  }
  mask[row * NCOLB + cb] = m;
}

// Single-WAVE sequential suppression scan. Mask rows are prefetched into an
// LDS double buffer with GLOBAL_LOAD_ASYNC_TO_LDS_B128 (ASYNCcnt-tracked),
// software-pipelined one row ahead: issue row i+1, wait ASYNCcnt<=2 so row i
// is resident, consume row i from LDS. Single-wave workgroup makes the
// per-iteration __syncthreads effectively free. Lane t owns remv words
// [t*8, t*8+8) and async-copies the matching 32B slice of each row.
__global__ void pd_nms_scan_kernel(const unsigned* __restrict__ mask,
                                   int* __restrict__ sel) {
  const int t = threadIdx.x;                  // 0..31, one wave
  __shared__ unsigned remv[NCOLB];            // 256-word suppression bitset
  __shared__ unsigned rowbuf[2][NCOLB];       // double-buffered mask rows
  __shared__ unsigned char keepf[NDET];

#pragma unroll
  for (int k = 0; k < 8; ++k) remv[t * 8 + k] = 0u;
  __syncthreads();

  // Async-copy the 32-byte slice of mask row i owned by this lane into
  // rowbuf[i&1]. Two b128 transfers; offset:16 applies to both LDS and
  // global sides per the ISA async pseudocode.
  const unsigned long long gbase = (unsigned long long)(uintptr_t)mask +
                                   (unsigned long long)(t * 32);
  const unsigned lbase0 = (unsigned)(uintptr_t)(&rowbuf[0][t * 8]);
  const unsigned lbase1 = (unsigned)(uintptr_t)(&rowbuf[1][t * 8]);

#define PD_ISSUE_ROW(i)                                                      \
  do {                                                                       \
    unsigned      _l = ((i) & 1) ? lbase1 : lbase0;                          \
    unsigned long long _g = gbase + (unsigned long long)(i) * (NCOLB * 4u);  \
    asm volatile("global_load_async_to_lds_b128 %0, %1, off\n\t"             \
                 "global_load_async_to_lds_b128 %0, %1, off offset:16"       \
                 :: "v"(_l), "v"(_g) : "memory");                            \
  } while (0)

  PD_ISSUE_ROW(0);

  for (int i = 0; i < NDET; ++i) {
    if (i + 1 < NDET) {
      PD_ISSUE_ROW(i + 1);
      asm volatile("s_wait_asynccnt 2" ::: "memory");   // row i landed
    } else {
      asm volatile("s_wait_asynccnt 0" ::: "memory");
    }
    __syncthreads();   // LDS visibility: prev OR + async row -> all lanes

    const bool alive = (((remv[i >> 5] >> (i & 31)) & 1u) == 0u);
    __syncthreads();   // read remv before anyone rewrites it

    if (alive) {
      const unsigned* rb = rowbuf[i & 1];
#pragma unroll
      for (int k = 0; k < 8; ++k) remv[t * 8 + k] |= rb[t * 8 + k];
    }
    if (t == 0) keepf[i] = (unsigned char)alive;
  }
  __syncthreads();
#undef PD_ISSUE_ROW

  if (t == 0) {
    int c = 0;
    for (int i = 0; i < NDET && c < TOPK; ++i) if (keepf[i])  sel[c++] = i;
    for (int i = 0; i < NDET && c < TOPK; ++i) if (!keepf[i]) sel[c++] = i;
  }
}

__global__ void pd_gather_kernel(const float* __restrict__ dets,
                                 const float* __restrict__ scores,
                                 const int* __restrict__ sidx,
                                 const int* __restrict__ sel,
                                 float* __restrict__ out) {
  const int r = blockIdx.x * blockDim.x + threadIdx.x;
  if (r >= TOPK) return;
  const int oi = sidx[sel[r]];
  out[r * 8 + 0] = scores[oi];
  out[r * 8 + 1] = dets[oi * 18 + 0];
  out[r * 8 + 2] = dets[oi * 18 + 1];
  out[r * 8 + 3] = dets[oi * 18 + 2];
  out[r * 8 + 4] = dets[oi * 18 + 4];
  out[r * 8 + 5] = dets[oi * 18 + 5];
  out[r * 8 + 6] = dets[oi * 18 + 8];
  out[r * 8 + 7] = dets[oi * 18 + 9];
}

extern "C" void kernel_launch(void* const* d_in, const int* in_sizes, int n_in,
                              void* d_out, int out_size, void* d_ws, size_t ws_size,
                              hipStream_t stream) {
  const float* x       = (const float*)d_in[0];   // (1, 8192, 1)
  const float* y       = (const float*)d_in[1];   // (1, 8192, 18)
  const float* anchors = (const float*)d_in[2];   // (8192, 2)
  float* out = (float*)d_out;                     // 100 x 8

  // Workspace layout (bytes), total ~9.34 MB:
  char* ws = (char*)d_ws;
  float*    dets   = (float*)(ws + 0);                       // 8192*18*4 = 589824
  float*    boxes  = (float*)(ws + 589824);                  // 8192*4*4  = 131072
  float*    scores = (float*)(ws + 720896);                  // 8192*4    = 32768
  int*      sidx   = (int*)(ws + 753664);                    // 8192*4    = 32768
  float*    sboxes = (float*)(ws + 786432);                  // 8192*4*4  = 131072
  unsigned* mask   = (unsigned*)(ws + 917504);               // 8192*256*4 = 8388608
  int*      sel    = (int*)(ws + 917504 + 8388608);          // 100*4

  // 1) decode (WMMA f32 16x16x4), boxes, sigmoid scores
  pd_prep_kernel<<<NDET / 16, 32, 0, stream>>>(x, y, anchors, dets, boxes, scores);

  // 2) bitonic sort by score desc (one workgroup, 64KB dynamic LDS)
  pd_sort_kernel<<<1, 1024, NDET * 8, stream>>>(scores, boxes, sidx, sboxes);

  // 3) pairwise suppression bitmask (wave32 blocks, 256x256 grid)
  dim3 mgrid(NCOLB, NCOLB);
  pd_nms_mask_kernel<<<mgrid, 32, 0, stream>>>(sboxes, mask);

  // 4) sequential suppression scan, async-pipelined mask-row prefetch
  pd_nms_scan_kernel<<<1, 32, 0, stream>>>(mask, sel);

  // 5) gather output rows
  pd_gather_kernel<<<1, 128, 0, stream>>>(dets, scores, sidx, sel, out);
}